// AngularPenaltySMLoss_32126355374446
// MI455X (gfx1250) — compile-verified
//
#include <hip/hip_runtime.h>
#include <hip/hip_bf16.h>
#include <stdint.h>

#define CLS   32000          // columns (C)
#define CLS4  (CLS / 4)      // float4 chunks per row
#define BLK   1024           // 32 waves of 32
#define NWAVE (BLK / 32)

__device__ __forceinline__ float wave_reduce_sum(float v) {
    // wave32: butterfly reduce
    #pragma unroll
    for (int off = 16; off > 0; off >>= 1)
        v += __shfl_xor(v, off, 32);
    return v;
}

// Block-wide sum for blockDim.x == 1024 (32 waves). `scr` needs >= 33 floats.
__device__ __forceinline__ float block_reduce_sum(float v, float* scr) {
    const int lane = threadIdx.x & 31;
    const int wid  = threadIdx.x >> 5;
    v = wave_reduce_sum(v);
    if (lane == 0) scr[wid] = v;
    __syncthreads();
    if (wid == 0) {
        float r = (lane < NWAVE) ? scr[lane] : 0.0f;
        r = wave_reduce_sum(r);
        if (lane == 0) scr[32] = r;
    }
    __syncthreads();
    float out = scr[32];
    __syncthreads();   // allow scr reuse by caller
    return out;
}

// One workgroup per row. Stage the 128KB row into LDS with CDNA5 async
// global->LDS B128 copies (ASYNCcnt), then both reduction passes read LDS.
__global__ void arcface_row_kernel(const float* __restrict__ x,
                                   const long long* __restrict__ target,
                                   float* __restrict__ row_loss) {
    extern __shared__ float smem[];          // [CLS] row + [64] scratch
    float* scr = smem + CLS;

    const int row = blockIdx.x;
    const float* rowp = x + (size_t)row * CLS;

    // ---- Phase 0: async stage row -> LDS (GVS form: saddr base + 32b voffset)
    for (int i = threadIdx.x; i < CLS4; i += BLK) {
        unsigned lds_off = (unsigned)(size_t)(smem + 4 * i);   // LDS byte addr (low 32b of generic ptr)
        unsigned voff    = (unsigned)(i * 16);                 // byte offset within row
        asm volatile("global_load_async_to_lds_b128 %0, %1, %2"
                     :: "v"(lds_off), "v"(voff), "s"(rowp)
                     : "memory");
    }
    asm volatile("s_wait_asynccnt 0" ::: "memory");
    __syncthreads();

    // ---- Phase 1: sum of squares (row L2 norm)
    const float4* l4 = (const float4*)smem;
    float ss = 0.0f;
    for (int i = threadIdx.x; i < CLS4; i += BLK) {
        float4 v = l4[i];
        ss += v.x * v.x + v.y * v.y + v.z * v.z + v.w * v.w;
    }
    const float sumsq  = block_reduce_sum(ss, scr);
    const float invn   = 1.0f / fmaxf(sqrtf(sumsq), 1e-12f);

    // ---- Phase 2: sum_j exp(S * xn_j)  ==  sum_j exp2(x_j * k)
    const float k = 30.0f * invn * 1.44269504088896340736f;    // S * invnorm * log2(e)
    float es = 0.0f;
    for (int i = threadIdx.x; i < CLS4; i += BLK) {
        float4 v = l4[i];
        es += __builtin_amdgcn_exp2f(v.x * k)
            + __builtin_amdgcn_exp2f(v.y * k)
            + __builtin_amdgcn_exp2f(v.z * k)
            + __builtin_amdgcn_exp2f(v.w * k);
    }
    const float row_sum = block_reduce_sum(es, scr);

    // ---- Phase 3: scalar tail (target column) on one lane
    if (threadIdx.x == 0) {
        const long long t = target[row];
        const float c  = smem[(int)t] * invn;                  // cos_t (unclamped)
        const float cc = fminf(fmaxf(c, -1.0f + 1e-7f), 1.0f - 1e-7f);
        // S * cos(acos(cc) + M) = S * (cc*cosM - sinM*sqrt(1-cc^2))
        const float num = 30.0f * (cc * 0.95533648912560601964f
                                   - 0.29552020666133957511f * sqrtf(fmaxf(1.0f - cc * cc, 0.0f)));
        const float L2E = 1.44269504088896340736f;
        const float denom = __builtin_amdgcn_exp2f(num * L2E) + row_sum
                          - __builtin_amdgcn_exp2f(30.0f * c * L2E);
        const float L = num - __builtin_amdgcn_logf(denom) * 0.69314718055994530942f;
        row_loss[row] = L;
    }
}

// Single-block deterministic reduction of the 4096 per-row losses -> -mean
__global__ void arcface_finalize_kernel(const float* __restrict__ row_loss,
                                        float* __restrict__ out, int n) {
    __shared__ float scr[64];
    float s = 0.0f;
    for (int i = threadIdx.x; i < n; i += BLK) s += row_loss[i];
    const float tot = block_reduce_sum(s, scr);
    if (threadIdx.x == 0) out[0] = -tot / (float)n;
}

extern "C" void kernel_launch(void* const* d_in, const int* in_sizes, int n_in,
                              void* d_out, int out_size, void* d_ws, size_t ws_size,
                              hipStream_t stream) {
    const float*     x      = (const float*)d_in[0];
    const long long* target = (const long long*)d_in[1];   // int64 targets
    float* row_loss = (float*)d_ws;                        // N floats scratch
    float* out      = (float*)d_out;

    const int N = in_sizes[1];                             // 4096 rows

    const size_t lds_bytes = (size_t)(CLS + 64) * sizeof(float);  // 128,256 B
    arcface_row_kernel<<<N, BLK, lds_bytes, stream>>>(x, target, row_loss);
    arcface_finalize_kernel<<<1, BLK, 0, stream>>>(row_loss, out, N);
}